// SparseRegionEnhanceLoss_23381801959585
// MI455X (gfx1250) — compile-verified
//
#include <hip/hip_runtime.h>
#include <math.h>

// Problem dims (fixed by the reference setup_inputs()).
#define BDIM 4
#define CDIM 21
#define HDIM 512
#define WDIM 512
#define HW   (HDIM * WDIM)          // 262144
#define NPIX (BDIM * HW)            // 1048576
#define PPT  4                      // pixels per thread in k_loss (b128 loads)
#define INV_LOG_501 0.16085950f     // 1 / ln(501)

typedef __attribute__((ext_vector_type(2))) float v2f;
typedef __attribute__((ext_vector_type(8))) float v8f;

// ---------------- union-find (ECL-CC style, deterministic final roots) ------

__device__ __forceinline__ unsigned find_root(const unsigned* __restrict__ L,
                                              unsigned a) {
  unsigned p = L[a];
  while (p != a) { a = p; p = L[a]; }
  return a;
}

__device__ __forceinline__ void unite(unsigned* __restrict__ L,
                                      unsigned a, unsigned b) {
  a = find_root(L, a);
  b = find_root(L, b);
  while (a != b) {
    if (a < b) { unsigned t = a; a = b; b = t; }   // a = larger index
    unsigned old = atomicMin(&L[a], b);            // hook larger root to smaller
    if (old == a) break;                           // we linked it
    a = old;                                       // someone else linked; follow
  }
}

// ---------------- kernels ---------------------------------------------------

__global__ void k_init(unsigned* __restrict__ label, unsigned* __restrict__ cnt) {
  unsigned p = blockIdx.x * blockDim.x + threadIdx.x;
  if (p < NPIX) { label[p] = p; cnt[p] = 0u; }
}

// One pass over right/down edges; atomic union-find handles transitivity.
__global__ void k_merge(const int* __restrict__ tgt, unsigned* __restrict__ label) {
  unsigned p = blockIdx.x * blockDim.x + threadIdx.x;
  if (p >= NPIX) return;
  int t = tgt[p];
  if (t <= 0) return;                        // background (class 0) excluded
  unsigned x = p & (WDIM - 1);
  unsigned y = (p >> 9) & (HDIM - 1);
  if (x + 1u < WDIM && tgt[p + 1] == t)     unite(label, p, p + 1);
  if (y + 1u < HDIM && tgt[p + WDIM] == t)  unite(label, p, p + WDIM);
}

// Flatten labels to final roots and count component sizes (integer atomics).
__global__ void k_count(const int* __restrict__ tgt,
                        unsigned* __restrict__ label,
                        unsigned* __restrict__ cnt) {
  unsigned p = blockIdx.x * blockDim.x + threadIdx.x;
  if (p >= NPIX) return;
  if (tgt[p] > 0) {
    unsigned r = find_root(label, p);
    label[p] = r;                 // path-compress for O(1) lookup in k_loss
    atomicAdd(&cnt[r], 1u);
  }
}

// region weight per pixel (reference: 1 unless fg and size < 500)
__device__ __forceinline__ float pix_weight(int t, unsigned lab,
                                            const unsigned* __restrict__ cnt) {
  float w = 1.0f;
  if (t > 0) {
    float sz = (float)cnt[lab];
    if (sz < 500.0f) w = 3.0f * __logf(sz + 1.0f) * INV_LOG_501;
  }
  return w;
}

// Fused CE + region-weight + deterministic reduction, 4 pixels/thread.
// Each wave VMEM request is 512 B contiguous (global_load_b128) per channel
// plane — instruction-efficient streaming of the 88 MB logits tensor.
// Wave-level reduction uses V_WMMA_F32_16X16X4_F32 with B = ones:
//   A (16x4 f32): lane m (m<16) holds {v_m, 0}; lane m+16 holds {v_{m+16}, 0}.
//   With all-ones B, D[M][N] = v_M + v_{M+16} — exact f32, fixed hardware
//   order, bit-deterministic.
__global__ void k_loss(const float* __restrict__ logits,
                       const int* __restrict__ tgt,
                       const unsigned* __restrict__ label,
                       const unsigned* __restrict__ cnt,
                       float* __restrict__ partial) {
  const unsigned tid = blockIdx.x * blockDim.x + threadIdx.x;  // exact cover
  const unsigned p4  = tid * PPT;                // first of 4 consecutive pixels
  const unsigned b   = p4 / HW;
  const unsigned off = p4 - b * HW;              // multiple of 4 -> 16B aligned
  const float* base  = logits + (size_t)b * (size_t)CDIM * HW + off;

  __builtin_prefetch(base, 0, 0);                // global_prefetch_b8

  const int4 t4 = *(const int4*)(tgt + p4);

  // log-softmax CE per component: ce = max + log(sum exp(l-max)) - l[target]
  float4 v[CDIM];
  float4 m;  m.x = m.y = m.z = m.w = -1e30f;
  float4 lt; lt.x = lt.y = lt.z = lt.w = 0.0f;
#pragma unroll
  for (int c = 0; c < CDIM; ++c) {
    float4 l = *(const float4*)(base + (size_t)c * HW);  // b128, coalesced
    v[c] = l;
    m.x = fmaxf(m.x, l.x); m.y = fmaxf(m.y, l.y);
    m.z = fmaxf(m.z, l.z); m.w = fmaxf(m.w, l.w);
    if (c == t4.x) lt.x = l.x;                  // v_cndmask, no dyn indexing
    if (c == t4.y) lt.y = l.y;
    if (c == t4.z) lt.z = l.z;
    if (c == t4.w) lt.w = l.w;
  }
  float4 s; s.x = s.y = s.z = s.w = 0.0f;
#pragma unroll
  for (int c = 0; c < CDIM; ++c) {
    s.x += __expf(v[c].x - m.x);
    s.y += __expf(v[c].y - m.y);
    s.z += __expf(v[c].z - m.z);
    s.w += __expf(v[c].w - m.w);
  }
  const float ce0 = __logf(s.x) + m.x - lt.x;
  const float ce1 = __logf(s.y) + m.y - lt.y;
  const float ce2 = __logf(s.z) + m.z - lt.z;
  const float ce3 = __logf(s.w) + m.w - lt.w;

  // dynamic sparse-region weights (gathers on 4 MB count table)
  const uint4 lab4 = *(const uint4*)(label + p4);
  const float w0 = pix_weight(t4.x, lab4.x, cnt);
  const float w1 = pix_weight(t4.y, lab4.y, cnt);
  const float w2 = pix_weight(t4.z, lab4.z, cnt);
  const float w3 = pix_weight(t4.w, lab4.w, cnt);

  const float val = (ce0 * w0 + ce1 * w1) + (ce2 * w2 + ce3 * w3);

  // ---- wave32 reduction via f32 WMMA (EXEC is all-ones here: exact cover,
  // all divergence above has reconverged) ----
  v2f a;  a[0] = val;  a[1] = 0.0f;
  v2f bb; bb[0] = 1.0f; bb[1] = 1.0f;           // B = all ones (layout-agnostic)
  v8f acc = {};
  acc = __builtin_amdgcn_wmma_f32_16x16x4_f32(
      /*neg_a=*/false, a, /*neg_b=*/false, bb,
      /*c_mod=*/(short)0, acc, /*reuse_a=*/false, /*reuse_b=*/false);

  // lane L<16 holds rows 0..7 sums in acc[0..7]; lane L>=16 rows 8..15
  float t8 = ((acc[0] + acc[1]) + (acc[2] + acc[3])) +
             ((acc[4] + acc[5]) + (acc[6] + acc[7]));
  float wsum = t8 + __shfl_xor(t8, 16, 32);     // full 32-lane sum, fixed order

  __shared__ float lds[8];                      // 256 threads = 8 waves
  const int lane = threadIdx.x & 31;
  const int wid  = threadIdx.x >> 5;
  if (lane == 0) lds[wid] = wsum;
  __syncthreads();
  if (threadIdx.x == 0) {
    float tot = 0.0f;
#pragma unroll
    for (int i = 0; i < 8; ++i) tot += lds[i];  // fixed order
    partial[blockIdx.x] = tot;
  }
}

// Deterministic final reduction of block partials, then mean.
__global__ void k_final(const float* __restrict__ partial,
                        float* __restrict__ out, int n) {
  __shared__ float sh[256];
  float s = 0.0f;
  for (int i = threadIdx.x; i < n; i += 256) s += partial[i];  // fixed order
  sh[threadIdx.x] = s;
  __syncthreads();
  for (int st = 128; st > 0; st >>= 1) {
    if ((int)threadIdx.x < st) sh[threadIdx.x] += sh[threadIdx.x + st];
    __syncthreads();
  }
  if (threadIdx.x == 0) out[0] = sh[0] * (1.0f / (float)NPIX);
}

// ---------------- launch ----------------------------------------------------

extern "C" void kernel_launch(void* const* d_in, const int* in_sizes, int n_in,
                              void* d_out, int out_size, void* d_ws, size_t ws_size,
                              hipStream_t stream) {
  (void)in_sizes; (void)n_in; (void)out_size; (void)ws_size;

  const float* logits = (const float*)d_in[0];
  const int*   tgt    = (const int*)d_in[1];
  float*       out    = (float*)d_out;

  // workspace: labels (4MB) | counts (4MB) | partials (4KB)
  unsigned* label   = (unsigned*)d_ws;
  unsigned* cnt     = label + NPIX;
  float*    partial = (float*)(cnt + NPIX);

  const int threads = 256;
  const int blocks  = NPIX / threads;            // 4096, exact cover
  const int blocksL = NPIX / (threads * PPT);    // 1024, exact cover

  k_init <<<blocks,  threads, 0, stream>>>(label, cnt);
  k_merge<<<blocks,  threads, 0, stream>>>(tgt, label);
  k_count<<<blocks,  threads, 0, stream>>>(tgt, label, cnt);
  k_loss <<<blocksL, threads, 0, stream>>>(logits, tgt, label, cnt, partial);
  k_final<<<1, 256, 0, stream>>>(partial, out, blocksL);
}